// InfoNCE_81853486727364
// MI455X (gfx1250) — compile-verified
//
#include <hip/hip_runtime.h>
#include <hip/hip_bf16.h>

typedef __attribute__((ext_vector_type(2))) float v2f;
typedef __attribute__((ext_vector_type(8))) float v8f;
typedef __attribute__((ext_vector_type(4))) unsigned int u32x4;
typedef __attribute__((ext_vector_type(4))) int i32x4;
typedef __attribute__((ext_vector_type(8))) int i32x8;

#define B_N 16384
#define D_K 128
#define TEMP_INV 10.0f        // 1 / 0.1
#define NCHUNK 4              // column splits (grid.y)
#define ROWS_PER_BLOCK 128    // 8 waves * 16 rows
#define COLS_PER_GROUP 64     // staging tile (64 rows x 128 floats)
#define LDS_STRIDE 132        // floats; 528B rows: 16B aligned, banks spread 4-apart
#define BUF_FLOATS (COLS_PER_GROUP * LDS_STRIDE)

// workspace layout (float offsets)
#define WS_BLOCKLOSS 0                        // [64]
#define WS_BLOCKCNT  64                       // [64]
#define WS_INV       128                      // [B_N]
#define WS_POS       (128 + B_N)              // [NCHUNK][B_N]
#define WS_NEG       (128 + B_N + NCHUNK*B_N) // [NCHUNK][B_N]

#if defined(__gfx1250__) && __has_builtin(__builtin_amdgcn_tensor_load_to_lds)
#define USE_TDM 1
#else
#define USE_TDM 0
#endif

// ---------------------------------------------------------------------------
// Kernel 1: per-row inverse norms (one wave per row)
// ---------------------------------------------------------------------------
__global__ void norm_kernel(const float* __restrict__ emb, float* __restrict__ ws) {
    const int wave = threadIdx.x >> 5;
    const int lane = threadIdx.x & 31;
    const int row  = blockIdx.x * 8 + wave;
    const float4 v = *(const float4*)(emb + row * D_K + lane * 4);
    float ss = v.x * v.x + v.y * v.y + v.z * v.z + v.w * v.w;
    ss += __shfl_xor(ss, 16);
    ss += __shfl_xor(ss, 8);
    ss += __shfl_xor(ss, 4);
    ss += __shfl_xor(ss, 2);
    ss += __shfl_xor(ss, 1);
    if (lane == 0)
        ws[WS_INV + row] = 1.0f / fmaxf(sqrtf(ss), 1e-12f);
}

// ---------------------------------------------------------------------------
// TDM issue helper: DMA 64 rows x 128 f32 (row stride 128) from global into
// LDS at byte offset ldsByteOff, padding +4 DWORDs after every 128 DWORDs
// (=> LDS row stride 132 floats). Wave-level instruction; caller guards to
// one wave per block. Tracked with TENSORcnt.
// ---------------------------------------------------------------------------
#if USE_TDM
__device__ __forceinline__ void tdm_load_group(const float* gsrc,
                                               unsigned ldsByteOff) {
    const unsigned long long ga = (unsigned long long)(uintptr_t)gsrc;
    u32x4 g0;
    g0.x = 1u;                                       // count=1, user mode
    g0.y = ldsByteOff;                               // lds_addr
    g0.z = (unsigned)(ga & 0xffffffffu);             // global_addr[31:0]
    g0.w = (unsigned)((ga >> 32) & 0x01ffffffu) | (2u << 30);  // addr[56:32] | type=2

    i32x8 g1;
    g1[0] = (2 << 16)        // data_size = 4 bytes
          | (1 << 20)        // pad_enable
          | (6 << 22)        // pad_interval: every 128 DWORDs
          | (3 << 25);       // pad_amount: 4 DWORDs
    g1[1] = (D_K & 0xffff) << 16;          // tensor_dim0 = 128 (bits [79:48] low half)
    g1[2] = (B_N & 0xffff) << 16;          // tensor_dim0 hi16=0 | tensor_dim1 lo16
    g1[3] = (D_K & 0xffff) << 16;          // tensor_dim1 hi16=0 | tile_dim0 = 128
    g1[4] = COLS_PER_GROUP;                // tile_dim1 = 64, tile_dim2 = 0
    g1[5] = D_K;                           // tensor_dim0_stride = 128 (low 32)
    g1[6] = 0;                             // stride hi | tensor_dim1_stride lo
    g1[7] = 0;

    const i32x4 z4 = {0, 0, 0, 0};
#if __has_include(<hip/amd_detail/amd_gfx1250_TDM.h>)
    const i32x8 z8 = {0, 0, 0, 0, 0, 0, 0, 0};
    __builtin_amdgcn_tensor_load_to_lds(g0, g1, z4, z4, z8, 0);   // clang-23 6-arg form
#else
    __builtin_amdgcn_tensor_load_to_lds(g0, g1, z4, z4, 0);       // ROCm 7.2 5-arg form
#endif
}
#endif

// ---------------------------------------------------------------------------
// Kernel 2: fused Gram-tile (WMMA fp32) + exp + masked pos/neg accumulation.
// Grid: (B_N/ROWS_PER_BLOCK, NCHUNK). Block: 256 threads = 8 waves.
// Column groups of 64 rows staged into LDS by the Tensor Data Mover,
// double-buffered so the next group's DMA overlaps this group's WMMA work.
// ---------------------------------------------------------------------------
__launch_bounds__(256, 1)
__global__ void sim_kernel(const float* __restrict__ emb,
                           const int*  __restrict__ labels,
                           const int*  __restrict__ mask,
                           float* __restrict__ ws) {
#if USE_TDM
    __shared__ float ldsB[2 * BUF_FLOATS];   // only LDS object -> offset 0
#else
    __shared__ float ldsB[BUF_FLOATS];
#endif

    const int wave = threadIdx.x >> 5;
    const int lane = threadIdx.x & 31;
    const int n    = lane & 15;   // column-within-tile / row-within-A-tile
    const int h    = lane >> 4;   // half-wave: selects K sub-pair (A/B), M-half (C/D)

    const int rowBase      = blockIdx.x * ROWS_PER_BLOCK + wave * 16;
    const int chunk        = blockIdx.y;
    const int colChunkBase = chunk * (B_N / NCHUNK);

    // --- A fragments: 16x4 fp32 layout, lane n = row n, half h = K pair (4s+2h, +1)
    v2f afrag[32];
    {
        const float* arow = emb + (rowBase + n) * D_K + 2 * h;
        #pragma unroll
        for (int s = 0; s < 32; ++s)
            afrag[s] = *(const v2f*)(arow + 4 * s);
    }

    // --- per-lane row metadata for the 8 output rows (M = 8*h + r)
    float invmT[8]; float validF[8]; int labm[8];
    #pragma unroll
    for (int r = 0; r < 8; ++r) {
        const int row = rowBase + 8 * h + r;
        invmT[r]  = ws[WS_INV + row] * TEMP_INV;
        labm[r]   = labels[row];
        validF[r] = (mask[row] != 0) ? 1.0f : 0.0f;
    }

    float posAcc[8], negAcc[8];
    #pragma unroll
    for (int r = 0; r < 8; ++r) { posAcc[r] = 0.0f; negAcc[r] = 0.0f; }

    const int nGroups = (B_N / NCHUNK) / COLS_PER_GROUP;  // 64

#if USE_TDM
    // prologue: start DMA of group 0 into buffer 0
    if (wave == 0)
        tdm_load_group(emb + (size_t)colChunkBase * D_K, 0u);
#endif

    for (int g = 0; g < nGroups; ++g) {
        const int colBase = colChunkBase + g * COLS_PER_GROUP;

#if USE_TDM
        const int cur = g & 1;
        __builtin_amdgcn_s_wait_tensorcnt(0);       // no-op for non-issuing waves
        __syncthreads();                            // data visible; buf[1-cur] free
        // The TDM write to ldsB is invisible to alias analysis; without this
        // compiler-level clobber the B-fragment loads fold to undef and the
        // WMMA feed gets CSE'd away (observed in an earlier build).
        asm volatile("" : : "r"(&ldsB[0]) : "memory");
        if (g + 1 < nGroups) {
            if (wave == 0)
                tdm_load_group(emb + (size_t)(colBase + COLS_PER_GROUP) * D_K,
                               (unsigned)((1 - cur) * BUF_FLOATS * 4));
        }
        const float* bufB = &ldsB[cur * BUF_FLOATS];
#else
        __syncthreads();
        #pragma unroll
        for (int i = 0; i < 8; ++i) {
            const int f  = threadIdx.x + i * 256;  // float4 index 0..2047
            const int cr = f >> 5;
            const int c4 = f & 31;
            const float4 v = *(const float4*)(emb + (colBase + cr) * D_K + c4 * 4);
            *(float4*)&ldsB[cr * LDS_STRIDE + c4 * 4] = v;
        }
        __syncthreads();
        const float* bufB = &ldsB[0];
#endif

        #pragma unroll
        for (int ct = 0; ct < 4; ++ct) {
            // B fragments: lane n = column n (row colBase+ct*16+n of E), half h = K pair.
            // Preload all 32 fragments into independent registers, then fence
            // the scheduler so the ds_load clause stays ahead of the WMMA
            // chain (otherwise the post-RA scheduler recycles one 4-VGPR
            // buffer with a full s_wait_dscnt 0x0 stall per WMMA pair).
            const float* brow = &bufB[(ct * 16 + n) * LDS_STRIDE + 2 * h];
            v2f bfrag[32];
            #pragma unroll
            for (int s = 0; s < 32; ++s)
                bfrag[s] = *(const v2f*)(brow + 4 * s);

#if __has_builtin(__builtin_amdgcn_sched_barrier)
            __builtin_amdgcn_sched_barrier(0);   // loads above, WMMAs below
#endif

            v8f acc = {};
            #pragma unroll
            for (int s = 0; s < 32; ++s)
                acc = __builtin_amdgcn_wmma_f32_16x16x4_f32(
                    false, afrag[s], false, bfrag[s], (short)0, acc, false, false);

            const int   coln      = colBase + ct * 16 + n;
            const float invc      = ws[WS_INV + coln];
            const int   labc      = labels[coln];
            const float colValidF = (mask[coln] != 0) ? 1.0f : 0.0f;

            // Hot loop: diagonal intentionally included in posAcc (on the
            // diagonal labels always match and mask is valid).
            #pragma unroll
            for (int r = 0; r < 8; ++r) {
                const float e  = __expf(acc[r] * (invmT[r] * invc));
                const float ev = e * (validF[r] * colValidF);
                const bool same = (labm[r] == labc);
                posAcc[r] += same ? ev : 0.0f;
                negAcc[r] += same ? 0.0f : ev;
            }

            // Wave-uniform, rare branch: the one tile containing the diagonal
            // of this wave's 16-row stripe. Subtract the identical recomputed
            // term (CSE makes it bit-exact vs. what was added above).
            if (colBase + ct * 16 == rowBase) {
                #pragma unroll
                for (int r = 0; r < 8; ++r) {
                    const float e  = __expf(acc[r] * (invmT[r] * invc));
                    const float ev = e * (validF[r] * colValidF);
                    posAcc[r] -= (n == 8 * h + r) ? ev : 0.0f;
                }
            }
        }
    }

    // reduce over the 16 lanes of each half-wave (bits 0..3; h bit untouched),
    // then lane n==r stores this (row, chunk) partial — unique owner, no atomics.
    #pragma unroll
    for (int r = 0; r < 8; ++r) {
        float p = posAcc[r], q = negAcc[r];
        p += __shfl_xor(p, 1); p += __shfl_xor(p, 2);
        p += __shfl_xor(p, 4); p += __shfl_xor(p, 8);
        q += __shfl_xor(q, 1); q += __shfl_xor(q, 2);
        q += __shfl_xor(q, 4); q += __shfl_xor(q, 8);
        if (n == r) {
            const int row = rowBase + 8 * h + r;
            ws[WS_POS + chunk * B_N + row] = p;
            ws[WS_NEG + chunk * B_N + row] = q;
        }
    }
}

// ---------------------------------------------------------------------------
// Kernel 3: per-row loss + deterministic block reduction (64 blocks x 256)
// ---------------------------------------------------------------------------
__global__ void loss_kernel(const int* __restrict__ mask, float* __restrict__ ws) {
    const int i = blockIdx.x * 256 + threadIdx.x;
    float pos = 0.0f, neg = 0.0f;
    #pragma unroll
    for (int c = 0; c < NCHUNK; ++c) {
        pos += ws[WS_POS + c * B_N + i];
        neg += ws[WS_NEG + c * B_N + i];
    }
    // exp(sim) > 0 always (sim in [-10,10] -> no underflow), so pos>0 <=> has_pos
    const bool rv  = (mask[i] != 0) && (pos > 0.0f);
    const float ls = rv ? -logf(pos / (pos + neg + 1e-8f)) : 0.0f;
    const float ct = rv ? 1.0f : 0.0f;

    float l = ls, c = ct;
    l += __shfl_xor(l, 16); c += __shfl_xor(c, 16);
    l += __shfl_xor(l, 8);  c += __shfl_xor(c, 8);
    l += __shfl_xor(l, 4);  c += __shfl_xor(c, 4);
    l += __shfl_xor(l, 2);  c += __shfl_xor(c, 2);
    l += __shfl_xor(l, 1);  c += __shfl_xor(c, 1);

    __shared__ float sl[8], sc[8];
    const int wave = threadIdx.x >> 5;
    const int lane = threadIdx.x & 31;
    if (lane == 0) { sl[wave] = l; sc[wave] = c; }
    __syncthreads();
    if (threadIdx.x == 0) {
        float L = 0.0f, C = 0.0f;
        #pragma unroll
        for (int w = 0; w < 8; ++w) { L += sl[w]; C += sc[w]; }
        ws[WS_BLOCKLOSS + blockIdx.x] = L;
        ws[WS_BLOCKCNT  + blockIdx.x] = C;
    }
}

// ---------------------------------------------------------------------------
// Kernel 4: serial finalize (deterministic)
// ---------------------------------------------------------------------------
__global__ void final_kernel(const float* __restrict__ ws, float* __restrict__ out) {
    if (threadIdx.x == 0) {
        float L = 0.0f, C = 0.0f;
        for (int i = 0; i < 64; ++i) {
            L += ws[WS_BLOCKLOSS + i];
            C += ws[WS_BLOCKCNT + i];
        }
        out[0] = (C > 0.0f) ? (L / fmaxf(C, 1.0f)) : 0.0f;
    }
}

extern "C" void kernel_launch(void* const* d_in, const int* in_sizes, int n_in,
                              void* d_out, int out_size, void* d_ws, size_t ws_size,
                              hipStream_t stream) {
    const float* emb    = (const float*)d_in[0];
    const int*   labels = (const int*)d_in[1];
    const int*   mask   = (const int*)d_in[2];
    float*       ws     = (float*)d_ws;
    float*       out    = (float*)d_out;

    norm_kernel<<<dim3(B_N / 8), dim3(256), 0, stream>>>(emb, ws);
    sim_kernel<<<dim3(B_N / ROWS_PER_BLOCK, NCHUNK), dim3(256), 0, stream>>>(
        emb, labels, mask, ws);
    loss_kernel<<<dim3(B_N / 256), dim3(256), 0, stream>>>(mask, ws);
    final_kernel<<<1, 32, 0, stream>>>(ws, out);
}